// BiRNNLM_76510547411092
// MI455X (gfx1250) — compile-verified
//
#include <hip/hip_runtime.h>
#include <math.h>

// ---------------------------------------------------------------------------
// BiRNN LM for MI455X (gfx1250, wave32).
//
// Roofline: output = 4096 x 32000 f32 = 512 MB  ->  ~22 us at 23.3 TB/s.
// Projection FLOPs (~8.4 GF with recompute) are trivial next to that, so the
// plan is: (1) tiny serial scan kernel, (2) WMMA pass computing online-softmax
// stats without writing logits, (3) WMMA pass recomputing logits and writing
// log_softmax once with non-temporal stores. V (2 MB) and total_h (256 KB)
// stay resident in the 192 MB L2 across both passes.
//
// GEMM uses exact-f32 V_WMMA_F32_16X16X4_F32 chained 4x for K=16.
// ---------------------------------------------------------------------------

#define VOCAB  32000
#define EMB    32
#define HH     8
#define SEQ    256
#define BATCH  16
#define ROWS   (SEQ * BATCH)     // 4096
#define TWOH   16
#define VSPLIT 4
#define VCHUNK (VOCAB / VSPLIT)  // 8000
#define NTILES (VCHUNK / 16)     // 500 16-wide vocab tiles per split

typedef __attribute__((ext_vector_type(2))) float v2f;
typedef __attribute__((ext_vector_type(8))) float v8f;

// ---------------------------------------------------------------------------
// Kernel 1: the two recurrent scans. Block 0 = forward, block 1 = backward.
// 128 threads = (batch b, unit j); h state + embedding tile live in LDS.
// Emits the PRE-update hidden state (matches jax.lax.scan carry-out order).
// total_h layout: th[(t*BATCH+b)*16 + j] fwd in cols 0..7, bwd in cols 8..15.
// ---------------------------------------------------------------------------
__global__ __launch_bounds__(128)
void rnn_scan_kernel(const int* __restrict__ idx,
                     const float* __restrict__ emb,
                     const float* __restrict__ U_f, const float* __restrict__ W_f,
                     const float* __restrict__ U_b, const float* __restrict__ W_b,
                     const float* __restrict__ b1f, const float* __restrict__ b2f,
                     const float* __restrict__ b1b, const float* __restrict__ b2b,
                     float* __restrict__ th)
{
    const bool bwd = (blockIdx.x == 1);
    const int  tid = threadIdx.x;
    const int  b   = tid >> 3;   // batch 0..15
    const int  j   = tid & 7;    // hidden unit 0..7

    const float* U = bwd ? U_b : U_f;
    const float* W = bwd ? W_b : W_f;
    const float  bias = bwd ? (b1b[j] + b2b[j]) : (b1f[j] + b2f[j]);

    float u[EMB];
#pragma unroll
    for (int k = 0; k < EMB; ++k) u[k] = U[j * EMB + k];
    float w[HH];
#pragma unroll
    for (int k = 0; k < HH; ++k) w[k] = W[j * HH + k];

    __shared__ float hs[BATCH][HH];
    __shared__ float es[BATCH][EMB];
    hs[b][j] = 0.0f;
    __syncthreads();

    const int eoff = (tid & 7) * 4;  // 4 floats of own batch row per thread

    for (int step = 0; step < SEQ; ++step) {
        const int t = bwd ? (SEQ - 1 - step) : step;

        // emit pre-update state
        th[(t * BATCH + b) * TWOH + (bwd ? (HH + j) : j)] = hs[b][j];

        // snapshot h for this batch row
        float hv[HH];
#pragma unroll
        for (int k = 0; k < HH; ++k) hv[k] = hs[b][k];

        // cooperative embedding-row gather into LDS
        const long row = (long)idx[t * BATCH + b];
        const float4 ev = *(const float4*)(emb + row * EMB + eoff);
        es[b][eoff + 0] = ev.x; es[b][eoff + 1] = ev.y;
        es[b][eoff + 2] = ev.z; es[b][eoff + 3] = ev.w;
        __syncthreads();

        float acc = bias;
#pragma unroll
        for (int k = 0; k < EMB; ++k) acc += u[k] * es[b][k];
#pragma unroll
        for (int k = 0; k < HH; ++k) acc += w[k] * hv[k];

        hs[b][j] = tanhf(acc);
        __syncthreads();
    }
}

// online-softmax merge: (m,s) <- combine((m,s),(m2,s2))
__device__ inline void osm_merge(float& m, float& s, float m2, float s2) {
    float mn = fmaxf(m, m2);
    s = s * __expf(m - mn) + s2 * __expf(m2 - mn);
    m = mn;
}

// Load the four K=4 A/B fragments for the chained f32 WMMAs.
// ISA layout: lanes 0-15 hold K={4i,4i+1}, lanes 16-31 hold K={4i+2,4i+3};
// row index = lane & 15. Same pattern for A (rows of total_h tile) and B
// (rows of V, since B[k][n] = V[vj+n][k]).
__device__ inline void load_frag4(const float* __restrict__ base16,
                                  int row, int koff, v2f f[4]) {
#pragma unroll
    for (int i = 0; i < 4; ++i)
        f[i] = *(const v2f*)(base16 + (long)row * TWOH + 4 * i + koff);
}

__device__ inline v8f wmma_tile(const v2f a[4], const v2f b[4]) {
    v8f acc = {};
#pragma unroll
    for (int i = 0; i < 4; ++i)
        acc = __builtin_amdgcn_wmma_f32_16x16x4_f32(
            /*neg_a=*/false, a[i], /*neg_b=*/false, b[i],
            /*c_mod=*/(short)0, acc, /*reuse_a=*/false, /*reuse_b=*/false);
    return acc;
}

// ---------------------------------------------------------------------------
// Kernel 2 (pass 1): per-row online-softmax stats over one vocab split.
// grid = (ROWS/16, VSPLIT), block = 256 (8 waves); wave w owns tiles w, w+8,..
// C/D layout: lane<16 reg r -> (M=r, N=lane); lane>=16 reg r -> (M=8+r, N=lane-16).
// ---------------------------------------------------------------------------
__global__ __launch_bounds__(256)
void logits_pass1_kernel(const float* __restrict__ th,
                         const float* __restrict__ V,
                         float* __restrict__ pm, float* __restrict__ ps)
{
    const int g     = blockIdx.x;
    const int split = blockIdx.y;
    const int lane  = threadIdx.x & 31;
    const int wave  = threadIdx.x >> 5;
    const int n     = lane & 15;
    const int koff  = (lane < 16) ? 0 : 2;
    const int rowbase = g * 16;

    v2f a[4];
    load_frag4(th, rowbase + n, koff, a);

    float mrun[8], srun[8];
#pragma unroll
    for (int r = 0; r < 8; ++r) { mrun[r] = -INFINITY; srun[r] = 0.0f; }

    for (int j = wave; j < NTILES; j += 8) {
        const int vj = split * VCHUNK + j * 16;
        v2f bf[4];
        load_frag4(V, vj + n, koff, bf);
        v8f acc = wmma_tile(a, bf);
#pragma unroll
        for (int r = 0; r < 8; ++r) {
            float x  = acc[r];
            float mn = fmaxf(mrun[r], x);
            srun[r]  = srun[r] * __expf(mrun[r] - mn) + __expf(x - mn);
            mrun[r]  = mn;
        }
    }

    // reduce across the 16 lanes sharing each output row (xor stays in-half)
#pragma unroll
    for (int off = 1; off < 16; off <<= 1) {
#pragma unroll
        for (int r = 0; r < 8; ++r) {
            float m2 = __shfl_xor(mrun[r], off, 32);
            float s2 = __shfl_xor(srun[r], off, 32);
            osm_merge(mrun[r], srun[r], m2, s2);
        }
    }

    __shared__ float red_m[8][16], red_s[8][16];
    if (lane == 0) {
#pragma unroll
        for (int r = 0; r < 8; ++r) { red_m[wave][r] = mrun[r]; red_s[wave][r] = srun[r]; }
    } else if (lane == 16) {
#pragma unroll
        for (int r = 0; r < 8; ++r) { red_m[wave][8 + r] = mrun[r]; red_s[wave][8 + r] = srun[r]; }
    }
    __syncthreads();

    if (threadIdx.x < 16) {
        float m = red_m[0][threadIdx.x], s = red_s[0][threadIdx.x];
        for (int wv = 1; wv < 8; ++wv)
            osm_merge(m, s, red_m[wv][threadIdx.x], red_s[wv][threadIdx.x]);
        pm[split * ROWS + rowbase + threadIdx.x] = m;
        ps[split * ROWS + rowbase + threadIdx.x] = s;
    }
}

// ---------------------------------------------------------------------------
// Kernel 3 (pass 2): combine split partials -> logsumexp, recompute logits
// (identical WMMA chain -> bit-identical x), write x - lse non-temporally.
// ---------------------------------------------------------------------------
__global__ __launch_bounds__(256)
void logits_pass2_kernel(const float* __restrict__ th,
                         const float* __restrict__ V,
                         const float* __restrict__ pm, const float* __restrict__ ps,
                         float* __restrict__ out)
{
    const int g     = blockIdx.x;
    const int split = blockIdx.y;
    const int lane  = threadIdx.x & 31;
    const int wave  = threadIdx.x >> 5;
    const int n     = lane & 15;
    const int koff  = (lane < 16) ? 0 : 2;
    const int rowbase = g * 16;

    __shared__ float lse[16];
    if (threadIdx.x < 16) {
        const int row = rowbase + threadIdx.x;
        float m = pm[row], s = ps[row];
        for (int k = 1; k < VSPLIT; ++k)
            osm_merge(m, s, pm[k * ROWS + row], ps[k * ROWS + row]);
        lse[threadIdx.x] = m + logf(s);
    }
    __syncthreads();

    v2f a[4];
    load_frag4(th, rowbase + n, koff, a);

    const int mbase = (lane < 16) ? 0 : 8;
    float bias[8];
#pragma unroll
    for (int r = 0; r < 8; ++r) bias[r] = lse[mbase + r];

    for (int j = wave; j < NTILES; j += 8) {
        const int vj = split * VCHUNK + j * 16;
        v2f bf[4];
        load_frag4(V, vj + n, koff, bf);
        v8f acc = wmma_tile(a, bf);
#pragma unroll
        for (int r = 0; r < 8; ++r) {
            const long row = rowbase + mbase + r;
            __builtin_nontemporal_store(acc[r] - bias[r],
                                        out + row * (long)VOCAB + vj + n);
        }
    }
}

// ---------------------------------------------------------------------------
extern "C" void kernel_launch(void* const* d_in, const int* in_sizes, int n_in,
                              void* d_out, int out_size, void* d_ws, size_t ws_size,
                              hipStream_t stream) {
    const int*   idx = (const int*)  d_in[0];
    const float* emb = (const float*)d_in[1];
    const float* U_f = (const float*)d_in[2];
    const float* W_f = (const float*)d_in[3];
    const float* U_b = (const float*)d_in[4];
    const float* W_b = (const float*)d_in[5];
    const float* V   = (const float*)d_in[6];
    const float* b1f = (const float*)d_in[7];
    const float* b2f = (const float*)d_in[8];
    const float* b1b = (const float*)d_in[9];
    const float* b2b = (const float*)d_in[10];
    float* out = (float*)d_out;

    // workspace: total_h (4096*16) | partial m (4*4096) | partial s (4*4096)
    float* th = (float*)d_ws;
    float* pm = th + ROWS * TWOH;
    float* ps = pm + VSPLIT * ROWS;

    rnn_scan_kernel<<<dim3(2), dim3(128), 0, stream>>>(
        idx, emb, U_f, W_f, U_b, W_b, b1f, b2f, b1b, b2b, th);

    dim3 grid(ROWS / 16, VSPLIT);
    logits_pass1_kernel<<<grid, dim3(256), 0, stream>>>(th, V, pm, ps);
    logits_pass2_kernel<<<grid, dim3(256), 0, stream>>>(th, V, pm, ps, out);
}